// EffectAlgebraBank_58574763982848
// MI455X (gfx1250) — compile-verified
//
#include <hip/hip_runtime.h>
#include <hip/hip_bf16.h>
#include <math.h>

// Problem constants (from reference)
#define B_   128
#define H_   8
#define D_   256
#define M_   8192
#define BH_  (B_ * H_)
#define RANK_ 16
#define NEG_INF (-3.402823466e+38f)

typedef __attribute__((ext_vector_type(2))) float v2f;
typedef __attribute__((ext_vector_type(8))) float v8f;

// ---------------------------------------------------------------------------
// Reductions (wave32: __shfl_xor over 32 lanes, then LDS across 8 waves)
// ---------------------------------------------------------------------------
__device__ inline void block_reduce2(float& a, float& b, float* lds /*>=16 floats*/) {
    const int lane = threadIdx.x & 31;
    const int w    = threadIdx.x >> 5;
    #pragma unroll
    for (int off = 16; off > 0; off >>= 1) {
        a += __shfl_xor(a, off, 32);
        b += __shfl_xor(b, off, 32);
    }
    if (lane == 0) { lds[w] = a; lds[8 + w] = b; }
    __syncthreads();
    float sa = 0.f, sb = 0.f;
    #pragma unroll
    for (int i = 0; i < 8; ++i) { sa += lds[i]; sb += lds[8 + i]; }
    __syncthreads();
    a = sa; b = sb;
}

// ---------------------------------------------------------------------------
// Kernel 1: q[b,h,i] = sum_j (Pr[h,i,j] + i*Pi[h,i,j]) * (pr[b,j] + i*pi[b,j])
// One block per (b,h); thread t computes row i = t. psi row cached in LDS.
// ---------------------------------------------------------------------------
__global__ __launch_bounds__(256)
void eab_probe_kernel(const float* __restrict__ psi,
                      const float* __restrict__ probe_r,
                      const float* __restrict__ probe_i,
                      float* __restrict__ qr, float* __restrict__ qi) {
    const int b = blockIdx.x, h = blockIdx.y, t = threadIdx.x;
    __shared__ float spr[D_], spi[D_];
    spr[t] = psi[((size_t)b * D_ + t) * 2 + 0];
    spi[t] = psi[((size_t)b * D_ + t) * 2 + 1];
    __syncthreads();
    const float* Pr = probe_r + ((size_t)h * D_ + t) * D_;
    const float* Pi = probe_i + ((size_t)h * D_ + t) * D_;
    float qrv = 0.f, qiv = 0.f;
    #pragma unroll 4
    for (int j = 0; j < D_; ++j) {
        const float prj = Pr[j], pij = Pi[j];
        qrv = fmaf(prj, spr[j], fmaf(-pij, spi[j], qrv));
        qiv = fmaf(prj, spi[j], fmaf( pij, spr[j], qiv));
    }
    const size_t bh = (size_t)b * H_ + h;
    qr[bh * D_ + t] = qrv;
    qi[bh * D_ + t] = qiv;
}

// ---------------------------------------------------------------------------
// Kernel 2: row-normalize u (one block per m)
// ---------------------------------------------------------------------------
__global__ __launch_bounds__(256)
void eab_unorm_kernel(const float* __restrict__ u_real,
                      const float* __restrict__ u_imag,
                      float* __restrict__ urn, float* __restrict__ uin) {
    const int m = blockIdx.x, t = threadIdx.x;
    __shared__ float red[16];
    const size_t o = (size_t)m * D_ + t;
    const float a = u_real[o], b = u_imag[o];
    float ss = a * a + b * b, dummy = 0.f;
    block_reduce2(ss, dummy, red);
    const float inv = rsqrtf(fmaxf(ss, 1e-8f));
    urn[o] = a * inv;
    uin[o] = b * inv;
}

// ---------------------------------------------------------------------------
// Kernel 3: scores[bh, m] = sigmoid(s[m]) * |conj(u_m) . q_bh|^2 via f32 WMMA.
// Wave = one 16(m) x 16(bh) tile; block = 8 waves along m; grid (BH/16, M/128).
// A fragment (16x4 f32): lane L holds A[m0 + L%16][k + 2*(L/16) + {0,1}]
// B fragment (4x16 f32): lane L holds B[k + 2*(L/16) + {0,1}][bh0 + L%16]
//                        = q[bh0 + L%16][k + 2*(L/16) + {0,1}]  (contiguous)
// C/D (8 VGPRs): c[r] in lane L -> m = m0 + r + 8*(L/16), n = bh0 + L%16
// ---------------------------------------------------------------------------
__global__ __launch_bounds__(256)
void eab_scores_wmma_kernel(const float* __restrict__ ur,
                            const float* __restrict__ ui,
                            const float* __restrict__ qr,
                            const float* __restrict__ qi,
                            const float* __restrict__ s_logit,
                            float* __restrict__ scores) {
    const int lane = threadIdx.x & 31;
    const int wave = threadIdx.x >> 5;
    const int kh   = lane >> 4;        // 0 or 1 (lane half)
    const int l16  = lane & 15;
    const int bh0  = blockIdx.x * 16;
    const int m0   = (blockIdx.y * 8 + wave) * 16;

    const float* __restrict__ urow  = ur + (size_t)(m0 + l16) * D_;
    const float* __restrict__ uirow = ui + (size_t)(m0 + l16) * D_;
    const float* __restrict__ qrrow = qr + (size_t)(bh0 + l16) * D_;
    const float* __restrict__ qirow = qi + (size_t)(bh0 + l16) * D_;

    v8f accR = {}; // ip_r = ur.qr + ui.qi
    v8f accI = {}; // ip_i = ur.qi - ui.qr

    for (int k = 0; k < D_; k += 4) {
        __builtin_prefetch(urow  + k + 64, 0, 0);
        __builtin_prefetch(qrrow + k + 64, 0, 0);
        const int ko = k + 2 * kh;
        const v2f aR = *(const v2f*)(urow  + ko);
        const v2f aI = *(const v2f*)(uirow + ko);
        const v2f bR = *(const v2f*)(qrrow + ko);
        const v2f bI = *(const v2f*)(qirow + ko);
        const v2f aIn = -aI; // f32 WMMA has no A/B NEG modifier -> negate in VALU

        accR = __builtin_amdgcn_wmma_f32_16x16x4_f32(false, aR,  false, bR, (short)0, accR, false, false);
        accR = __builtin_amdgcn_wmma_f32_16x16x4_f32(false, aI,  false, bI, (short)0, accR, false, false);
        accI = __builtin_amdgcn_wmma_f32_16x16x4_f32(false, aR,  false, bI, (short)0, accI, false, false);
        accI = __builtin_amdgcn_wmma_f32_16x16x4_f32(false, aIn, false, bR, (short)0, accI, false, false);
    }

    const int n = bh0 + l16;
    #pragma unroll
    for (int r = 0; r < 8; ++r) {
        const int m  = m0 + r + 8 * kh;
        const float sig = 1.f / (1.f + expf(-s_logit[m]));
        const float v = accR[r] * accR[r] + accI[r] * accI[r];
        scores[(size_t)n * M_ + m] = sig * v;
    }
}

// ---------------------------------------------------------------------------
// Kernel 4: per-bh top-16 (only the leading 16 of top_k(32) reach the output).
// Scores row staged in 32 KB LDS; 16 argmax sweeps with lowest-index tiebreak.
// ---------------------------------------------------------------------------
__global__ __launch_bounds__(256)
void eab_topk_kernel(float* __restrict__ scores,
                     int* __restrict__ idx_out,
                     float* __restrict__ wgt_out) {
    const int bh = blockIdx.x, t = threadIdx.x;
    const int lane = t & 31, w = t >> 5;
    __shared__ float row[M_];
    __shared__ float bv[8];
    __shared__ int   bi[8];

    for (int i = t; i < M_; i += 256) row[i] = scores[(size_t)bh * M_ + i];
    __syncthreads();

    for (int sel = 0; sel < RANK_; ++sel) {
        float best = NEG_INF; int besti = 0x7fffffff;
        for (int i = t; i < M_; i += 256) {
            const float v = row[i];
            if (v > best || (v == best && i < besti)) { best = v; besti = i; }
        }
        #pragma unroll
        for (int off = 16; off > 0; off >>= 1) {
            const float ov = __shfl_xor(best, off, 32);
            const int   oi = __shfl_xor(besti, off, 32);
            if (ov > best || (ov == best && oi < besti)) { best = ov; besti = oi; }
        }
        if (lane == 0) { bv[w] = best; bi[w] = besti; }
        __syncthreads();
        if (t == 0) {
            float gb = bv[0]; int gi = bi[0];
            #pragma unroll
            for (int i = 1; i < 8; ++i)
                if (bv[i] > gb || (bv[i] == gb && bi[i] < gi)) { gb = bv[i]; gi = bi[i]; }
            idx_out[bh * RANK_ + sel] = gi;
            wgt_out[bh * RANK_ + sel] = sqrtf(fmaxf(gb, 0.f));
            row[gi] = NEG_INF;
        }
        __syncthreads();
    }
}

// ---------------------------------------------------------------------------
// Kernel 5: per-bh complex Gram-Schmidt QR (256x16), diag(Q^H W), outputs.
// Columns live in LDS (32 KB); thread t owns element d = t of every column.
// ---------------------------------------------------------------------------
__global__ __launch_bounds__(256)
void eab_qr_kernel(const float* __restrict__ urn, const float* __restrict__ uin,
                   const float* __restrict__ w_real, const float* __restrict__ w_imag,
                   const int* __restrict__ topk_idx, const float* __restrict__ topk_w,
                   float* __restrict__ out) {
    const int bh = blockIdx.x, d = threadIdx.x;
    __shared__ float Qr[RANK_][D_];
    __shared__ float Qi[RANK_][D_];
    __shared__ int   sidx[RANK_];
    __shared__ float swgt[RANK_];
    __shared__ float red[16];

    if (d < RANK_) {
        sidx[d] = topk_idx[bh * RANK_ + d];
        swgt[d] = topk_w[bh * RANK_ + d];
    }
    __syncthreads();

    for (int c = 0; c < RANK_; ++c) {
        const size_t o = (size_t)sidx[c] * D_ + d;
        Qr[c][d] = urn[o] * swgt[c];
        Qi[c][d] = uin[o] * swgt[c];
    }
    __syncthreads();

    // Gram-Schmidt: leading 16 Q columns match reference QR's leading columns
    for (int c = 0; c < RANK_; ++c) {
        for (int p = 0; p < c; ++p) {
            // proj = q_p^H . u_c
            float prr = Qr[p][d] * Qr[c][d] + Qi[p][d] * Qi[c][d];
            float pri = Qr[p][d] * Qi[c][d] - Qi[p][d] * Qr[c][d];
            block_reduce2(prr, pri, red);
            // u_c -= q_p * proj
            const float nr = Qr[c][d] - (Qr[p][d] * prr - Qi[p][d] * pri);
            const float ni = Qi[c][d] - (Qr[p][d] * pri + Qi[p][d] * prr);
            __syncthreads();
            Qr[c][d] = nr; Qi[c][d] = ni;
            __syncthreads();
        }
        float nn = Qr[c][d] * Qr[c][d] + Qi[c][d] * Qi[c][d];
        float dummy = 0.f;
        block_reduce2(nn, dummy, red);
        const float inv = rsqrtf(fmaxf(nn, 1e-20f));
        const float nr = Qr[c][d] * inv, ni = Qi[c][d] * inv;
        __syncthreads();
        Qr[c][d] = nr; Qi[c][d] = ni;
        __syncthreads();
    }

    // diag_r = q_r^H . (w_c[idx_r] * wgt_r);  V[:,r] = q_r * diag_r
    const size_t Voff = (size_t)BH_ * D_ * RANK_ * 2;
    const size_t base = (((size_t)bh * D_ + d) * RANK_) * 2;
    for (int r = 0; r < RANK_; ++r) {
        const size_t o = (size_t)sidx[r] * D_ + d;
        const float wr = w_real[o] * swgt[r];
        const float wi = w_imag[o] * swgt[r];
        float drr = Qr[r][d] * wr + Qi[r][d] * wi;
        float dri = Qr[r][d] * wi - Qi[r][d] * wr;
        block_reduce2(drr, dri, red);
        const float vr = Qr[r][d] * drr - Qi[r][d] * dri;
        const float vi = Qr[r][d] * dri + Qi[r][d] * drr;
        out[base + 2 * r + 0] = Qr[r][d];
        out[base + 2 * r + 1] = Qi[r][d];
        out[Voff + base + 2 * r + 0] = vr;
        out[Voff + base + 2 * r + 1] = vi;
    }
}

// ---------------------------------------------------------------------------
// Launch
// ---------------------------------------------------------------------------
extern "C" void kernel_launch(void* const* d_in, const int* in_sizes, int n_in,
                              void* d_out, int out_size, void* d_ws, size_t ws_size,
                              hipStream_t stream) {
    const float* psi        = (const float*)d_in[0];
    const float* u_real     = (const float*)d_in[1];
    const float* u_imag     = (const float*)d_in[2];
    const float* w_real     = (const float*)d_in[3];
    const float* w_imag     = (const float*)d_in[4];
    const float* s_logit    = (const float*)d_in[5];
    const float* probe_real = (const float*)d_in[6];
    const float* probe_imag = (const float*)d_in[7];
    // d_in[8] = k (32), d_in[9] = rank (16): outputs depend only on top-16.

    // Workspace layout (~52.5 MB of floats)
    float* ws     = (float*)d_ws;
    float* qr     = ws;                          // BH*D
    float* qi     = qr  + (size_t)BH_ * D_;      // BH*D
    float* urn    = qi  + (size_t)BH_ * D_;      // M*D
    float* uin    = urn + (size_t)M_  * D_;      // M*D
    float* scores = uin + (size_t)M_  * D_;      // BH*M
    int*   tidx   = (int*)(scores + (size_t)BH_ * M_);   // BH*RANK
    float* twgt   = (float*)(tidx + (size_t)BH_ * RANK_); // BH*RANK

    eab_probe_kernel<<<dim3(B_, H_), 256, 0, stream>>>(psi, probe_real, probe_imag, qr, qi);
    eab_unorm_kernel<<<M_, 256, 0, stream>>>(u_real, u_imag, urn, uin);
    eab_scores_wmma_kernel<<<dim3(BH_ / 16, M_ / 128), 256, 0, stream>>>(urn, uin, qr, qi, s_logit, scores);
    eab_topk_kernel<<<BH_, 256, 0, stream>>>(scores, tidx, twgt);
    eab_qr_kernel<<<BH_, 256, 0, stream>>>(urn, uin, w_real, w_imag, tidx, twgt, (float*)d_out);
}